// basic_gcn_nc_11905649344937
// MI455X (gfx1250) — compile-verified
//
#include <hip/hip_runtime.h>

// -------- problem constants (match reference) --------
#define NN 100000      // nodes
#define NE 1600000     // edges
#define KDIM 128       // inner dim of every GEMM (DIMS in = 128 for all 3 layers)
#define LDS_PAD 144    // W row stride in LDS: 2*144 % 64banks == 32 -> conflict-free B-frag reads

typedef __attribute__((ext_vector_type(2))) float v2f;
typedef __attribute__((ext_vector_type(8))) float v8f;
typedef __attribute__((ext_vector_type(4))) unsigned int u32x4;
typedef __attribute__((ext_vector_type(4))) int i32x4;
typedef __attribute__((ext_vector_type(8))) int i32x8;

// ---------------- elementwise / graph-prep kernels ----------------

__global__ void zero_f32_kernel(float* __restrict__ p, int n) {
    int i = blockIdx.x * blockDim.x + threadIdx.x;
    if (i < n) p[i] = 0.0f;
}

__global__ void deg_accum_kernel(const long long* __restrict__ dst,
                                 const float* __restrict__ w,
                                 float* __restrict__ deg, int e_cnt) {
    int e = blockIdx.x * blockDim.x + threadIdx.x;
    if (e < e_cnt) unsafeAtomicAdd(&deg[(int)dst[e]], w[e]);
}

// deg -> deg^{-1/2} in place (self-loop weight 1 added; deg+1 > 0 always)
__global__ void rsqrt_inplace_kernel(float* __restrict__ deg, int n) {
    int i = blockIdx.x * blockDim.x + threadIdx.x;
    if (i < n) deg[i] = rsqrtf(deg[i] + 1.0f);
}

__global__ void edge_norm_kernel(const long long* __restrict__ src,
                                 const long long* __restrict__ dst,
                                 const float* __restrict__ w,
                                 const float* __restrict__ dinv,
                                 float* __restrict__ norm, int e_cnt) {
    int e = blockIdx.x * blockDim.x + threadIdx.x;
    if (e < e_cnt) norm[e] = dinv[(int)src[e]] * w[e] * dinv[(int)dst[e]];
}

// acc = dinv^2 * xw + bias  (self-loop message + bias; scatter adds on top)
__global__ void init_acc_kernel(float* __restrict__ acc,
                                const float* __restrict__ xw,
                                const float* __restrict__ dinv,
                                const float* __restrict__ bias,
                                int n_nodes, int F) {
    int i = blockIdx.x * blockDim.x + threadIdx.x;
    if (i < n_nodes * F) {
        int node = i / F, f = i - node * F;
        float di = dinv[node];
        acc[i] = di * di * xw[i] + bias[f];
    }
}

__global__ void relu_kernel(float* __restrict__ p, int n) {
    int i = blockIdx.x * blockDim.x + threadIdx.x;
    if (i < n) p[i] = fmaxf(p[i], 0.0f);
}

// ---------------- edge scatter: acc[dst] += norm * xw[src] ----------------
// 32 lanes (one wave) per edge; F/32 floats per lane; xw row (512B) is L2-resident.
template <int F>
__global__ void scatter_kernel(const float* __restrict__ xw,
                               const long long* __restrict__ src,
                               const long long* __restrict__ dst,
                               const float* __restrict__ norm,
                               float* __restrict__ acc) {
    long long gid = (long long)blockIdx.x * blockDim.x + threadIdx.x;
    int e = (int)(gid >> 5);
    int lane = (int)(gid & 31);
    if (e >= NE) return;
    constexpr int PER = F / 32;
    int s = (int)src[e], d = (int)dst[e];
    float nm = norm[e];
    const float* xr = xw + (size_t)s * F + lane * PER;
    float* ar = acc + (size_t)d * F + lane * PER;
    float v[PER];
#pragma unroll
    for (int j = 0; j < PER; ++j) v[j] = xr[j];   // vectorizes to b128/b64 load
#pragma unroll
    for (int j = 0; j < PER; ++j) unsafeAtomicAdd(&ar[j], nm * v[j]);
}

// ---------------- TDM: DMA W[128 x F] (row-major, dense) into LDS with padded rows ----------------
// D# per ISA ch.8: group0 = {count/type, lds_addr, global_addr}, group1 = dims/tile/pad,
// groups 2-3 zero (2D tensor). Hardware pads each F-dword row out to LDS_PAD dwords.
__device__ __forceinline__ void tdm_load_W(const float* W, float* Wlds, int F) {
    unsigned long long ga = (unsigned long long)(uintptr_t)W;
    // low 32 bits of a flat LDS pointer are the LDS byte offset (ISA 10.2 aperture mapping)
    unsigned lds_off = (unsigned)(uintptr_t)Wlds;

    u32x4 g0;
    g0[0] = 1u;                                         // count=1 valid, is_restore=0, no gather
    g0[1] = lds_off;                                    // lds_addr (bytes)
    g0[2] = (unsigned)(ga & 0xFFFFFFFFu);               // global_addr[31:0]
    g0[3] = (unsigned)((ga >> 32) & 0x01FFFFFFu) | (2u << 30);  // addr[56:32] | type=2

    // pad each row: F=128 -> every 2^7 dwords add 16 dwords (codes 6,15)
    //               F=64  -> every 2^6 dwords add 80 dwords (codes 5,79)
    unsigned padI = (F == 128) ? 6u : 5u;
    unsigned padA = (F == 128) ? 15u : 79u;

    i32x8 g1;
    g1[0] = (int)((2u << 16)            // data_size = 4 bytes
                | (1u << 20)            // pad_enable
                | (padI << 22) | (padA << 25));
    g1[1] = (int)((unsigned)F << 16);   // tensor_dim0[15:0] @ bits 63:48
    g1[2] = (int)(128u << 16);          // tensor_dim0 hi=0 | tensor_dim1[15:0]=128
    g1[3] = (int)((unsigned)F << 16);   // tensor_dim1 hi=0 | tile_dim0=F
    g1[4] = (int)128u;                  // tile_dim1=128 | tile_dim2=0
    g1[5] = (int)(unsigned)F;           // tensor_dim0_stride[31:0] = F (dense rows)
    g1[6] = 0;                          // stride hi | tensor_dim1_stride lo (unused, 2D)
    g1[7] = 0;
    i32x4 g2 = {0, 0, 0, 0};            // 2D: higher dims / iterate off
    i32x4 g3 = {0, 0, 0, 0};

#if __clang_major__ >= 23
    i32x8 g4 = {0, 0, 0, 0, 0, 0, 0, 0};
    __builtin_amdgcn_tensor_load_to_lds(g0, g1, g2, g3, g4, 0);
#else
    __builtin_amdgcn_tensor_load_to_lds(g0, g1, g2, g3, 0);
#endif
    __builtin_amdgcn_s_wait_tensorcnt(0);
}

// ---------------- fp32 WMMA GEMM: XW[nrows x F] = X[nrows x 128] @ W[128 x F] ----------------
// 256 threads = 8 waves; block covers 128 rows (16 rows per wave).
// A-frag layout (ISA 7.12.2, 32-bit A 16x4): lane<16 -> M=lane,K=k..k+1 ; lane>=16 -> M=lane-16,K=k+2..k+3
// B-frag mirrors with N across lanes; C/D: vgpr j -> M = j + 8*(lane/16), N = lane%16.
__global__ void gemm_wmma_f32_kernel(const float* __restrict__ X,
                                     const float* __restrict__ W,
                                     float* __restrict__ XW,
                                     int nrows, int F) {
    extern __shared__ float Wlds[];  // KDIM x LDS_PAD
    const int tid  = threadIdx.x;
    const int lane = tid & 31;
    const int wave = tid >> 5;
    const int l16  = lane & 15;
    const int lh   = lane >> 4;

    // wave 0 DMAs W into LDS via the Tensor Data Mover (EXEC-independent, wave-level op)
    if (wave == 0) tdm_load_W(W, Wlds, F);
    __syncthreads();

    const int r0   = blockIdx.x * 128 + wave * 16;
    const int row  = r0 + l16;
    const int rowc = row < nrows ? row : (nrows - 1);  // clamp: garbage rows only affect guarded stores
    const float* xr = X + (size_t)rowc * KDIM + 2 * lh;

    // preload full 16x128 A panel: 32 fragments of v2f (64 VGPRs)
    v2f a[32];
#pragma unroll
    for (int s = 0; s < 32; ++s) a[s] = *(const v2f*)(xr + 4 * s);

    const int ntiles = F >> 4;
    const float* wp = Wlds + 2 * lh * LDS_PAD + l16;
    for (int t = 0; t < ntiles; ++t) {
        const float* wpt = wp + t * 16;
        // preload all 32 B fragments (64 VGPRs): LDS loads pipeline among themselves,
        // then the 32 WMMAs run back-to-back on the accumulator chain with no LDS stalls.
        v2f b[32];
#pragma unroll
        for (int s = 0; s < 32; ++s) {
            v2f bb;
            bb.x = wpt[(4 * s) * LDS_PAD];
            bb.y = wpt[(4 * s + 1) * LDS_PAD];
            b[s] = bb;
        }
        v8f c = {};
#pragma unroll
        for (int s = 0; s < 32; ++s) {
            c = __builtin_amdgcn_wmma_f32_16x16x4_f32(
                    false, a[s], false, b[s], (short)0, c, false, false);
        }
#pragma unroll
        for (int j = 0; j < 8; ++j) {
            int orow = r0 + j + 8 * lh;
            if (orow < nrows) XW[(size_t)orow * F + t * 16 + l16] = c[j];
        }
    }
}

// ---------------- host-side orchestration ----------------

extern "C" void kernel_launch(void* const* d_in, const int* in_sizes, int n_in,
                              void* d_out, int out_size, void* d_ws, size_t ws_size,
                              hipStream_t stream) {
    const float*     x    = (const float*)d_in[0];
    const long long* ei   = (const long long*)d_in[1];   // int64 (2, E)
    const float*     ew   = (const float*)d_in[2];
    const float*     W1   = (const float*)d_in[3];
    const float*     b1   = (const float*)d_in[4];
    const float*     W2   = (const float*)d_in[5];
    const float*     b2   = (const float*)d_in[6];
    const float*     W3   = (const float*)d_in[7];
    const float*     b3   = (const float*)d_in[8];
    float*           out  = (float*)d_out;

    const long long* srcI = ei;
    const long long* dstI = ei + NE;

    // workspace layout (floats): dinv[N] | norm[E] | xw[N*128] | hbuf[N*128]  (~110 MB)
    float* ws   = (float*)d_ws;
    float* dinv = ws;
    float* norm = dinv + 100352;               // N rounded up
    float* xw   = norm + NE;
    float* hbuf = xw + (size_t)NN * 128;

    const int TPB = 256;
    const int gN    = (NN + TPB - 1) / TPB;
    const int gE    = (NE + TPB - 1) / TPB;
    const int gNF   = (NN * 128 + TPB - 1) / TPB;
    const int gNF64 = (NN * 64 + TPB - 1) / TPB;
    const int gScat = (int)(((long long)NE * 32 + TPB - 1) / TPB);
    const int gGemm = (NN + 127) / 128;
    const size_t ldsW = (size_t)KDIM * LDS_PAD * sizeof(float);  // 73728 B (< 320 KB WGP LDS)

    // graph prep: deg -> dinv -> per-edge norm (reused by all 3 layers)
    zero_f32_kernel<<<gN, TPB, 0, stream>>>(dinv, NN);
    deg_accum_kernel<<<gE, TPB, 0, stream>>>(dstI, ew, dinv, NE);
    rsqrt_inplace_kernel<<<gN, TPB, 0, stream>>>(dinv, NN);
    edge_norm_kernel<<<gE, TPB, 0, stream>>>(srcI, dstI, ew, dinv, norm, NE);

    // ---- layer 1: xw = x@W1 ; h = relu(scatter + dinv^2*xw + b1) ----
    gemm_wmma_f32_kernel<<<gGemm, TPB, ldsW, stream>>>(x, W1, xw, NN, 128);
    init_acc_kernel<<<gNF, TPB, 0, stream>>>(hbuf, xw, dinv, b1, NN, 128);
    scatter_kernel<128><<<gScat, TPB, 0, stream>>>(xw, srcI, dstI, norm, hbuf);
    relu_kernel<<<gNF, TPB, 0, stream>>>(hbuf, NN * 128);

    // ---- layer 2 (acc reuses hbuf: GEMM consumed it before init overwrites) ----
    gemm_wmma_f32_kernel<<<gGemm, TPB, ldsW, stream>>>(hbuf, W2, xw, NN, 128);
    init_acc_kernel<<<gNF, TPB, 0, stream>>>(hbuf, xw, dinv, b2, NN, 128);
    scatter_kernel<128><<<gScat, TPB, 0, stream>>>(xw, srcI, dstI, norm, hbuf);
    relu_kernel<<<gNF, TPB, 0, stream>>>(hbuf, NN * 128);

    // ---- layer 3: F = 64, no activation, accumulate directly into d_out ----
    gemm_wmma_f32_kernel<<<gGemm, TPB, ldsW, stream>>>(hbuf, W3, xw, NN, 64);
    init_acc_kernel<<<gNF64, TPB, 0, stream>>>(out, xw, dinv, b3, NN, 64);
    scatter_kernel<64><<<gScat, TPB, 0, stream>>>(xw, srcI, dstI, norm, out);
}